// OT_Loss_48335561949211
// MI455X (gfx1250) — compile-verified
//
#include <hip/hip_runtime.h>
#include <math.h>

// ---------------------------------------------------------------------------
// Entropic OT (Sinkhorn divergence) for MI455X / gfx1250.
//
// Compute-bound (~2.2e9 logsumexp terms, ~48KB input). All state lives in LDS.
// The cost matrix is recomputed on the fly with V_WMMA_F32_16X16X4_F32, and
// the whole LSE runs in the log2 domain so v_exp_f32 / v_log_f32 are used raw
// (no x*log2e multiply per exp):
//   A-row_i = [xc0*L/eps, xc1*L/eps, u2_i, 1]   (xc = x-128, L = log2 e)
//   B-col_j = [yc0,       yc1,       1,  sq2_j] (sq2 = -0.5|yc|^2 * L/eps)
//   D_ij    = u2_i + xc.yc*L/eps + sq2_j == log2e * (lw_i + (f_i - C_ij)/eps)
// One WMMA yields all 256 pre-exp2 LSE arguments of a 16x16 tile. The online
// softmax processes 4 tiles (64 rows) per update: 32 independent v_exp_f32
// stream through the trans pipe, and the loop-carried rescale exp happens
// once per 4 tiles. Fragments are staged in LDS in the exact register layout:
// one branchless ds_load_b64 per tile per lane (paired into ds_load_2addr).
// ---------------------------------------------------------------------------

typedef float v2f __attribute__((ext_vector_type(2)));
typedef float v8f __attribute__((ext_vector_type(8)));

#define NMAX     1024              // grid support points (32x32)
#define PPTS     256               // gt points per image
#define EPS_     0.0025f           // blur^2 = 0.05^2
#define LOG2E    1.4426950408889634f
#define LN2      0.6931471805599453f
#define IEL      (400.0f * LOG2E)  // log2e / eps
#define EPSLN2   (EPS_ * LN2)      // converts log2-LSE back to -eps*ln(...)
#define ITERS_   100
#define NEG_BIG  (-1.0e30f)        // stand-in for -inf (keeps WMMA NaN-free)

struct __align__(16) Smem {
    // Per-point WMMA fragment quads: {k0, k1, k2, k3}.
    // X side stores centered coords pre-scaled by log2e/eps; Y side raw.
    // Slots 2/3 are rewritten each half-step depending on row/col role.
    float fragX[4 * NMAX];
    float fragY[4 * NMAX];
    float lwX[NMAX], sqX[NMAX], f[NMAX];   // lw, sq already in log2 domain
    float lwY[NMAX], sqY[NMAX], g[NMAX];   // f, g in natural (eps) units
    float gfin[NMAX];
    float red[32];
};

__device__ __forceinline__ float max8(v8f d) {
    return fmaxf(fmaxf(fmaxf(d[0], d[1]), fmaxf(d[2], d[3])),
                 fmaxf(fmaxf(d[4], d[5]), fmaxf(d[6], d[7])));
}

// Balanced tree of 8 independent raw exp2s (trans pipe streams).
__device__ __forceinline__ float expsum8(v8f d, float mx) {
    float e0 = __builtin_amdgcn_exp2f(d[0] - mx);
    float e1 = __builtin_amdgcn_exp2f(d[1] - mx);
    float e2 = __builtin_amdgcn_exp2f(d[2] - mx);
    float e3 = __builtin_amdgcn_exp2f(d[3] - mx);
    float e4 = __builtin_amdgcn_exp2f(d[4] - mx);
    float e5 = __builtin_amdgcn_exp2f(d[5] - mx);
    float e6 = __builtin_amdgcn_exp2f(d[6] - mx);
    float e7 = __builtin_amdgcn_exp2f(d[7] - mx);
    return ((e0 + e1) + (e2 + e3)) + ((e4 + e5) + (e6 + e7));
}

// One Sinkhorn half-step: out[j] = -eps * LSE_i( rlw_i + (rf_i - C_ij)/eps )
// computed as -eps*ln2 * LSE2_i( D_ij ). Ends with a barrier.
__device__ __forceinline__ void half_step(
    float* __restrict__ rowFrag, const float* __restrict__ rlw,
    const float* __restrict__ rsq, const float* __restrict__ rf,
    float* __restrict__ colFrag, const float* __restrict__ csq,
    float* __restrict__ outPot, int n, int m)
{
    const int tid = threadIdx.x;

    // Refresh role-dependent fragment slots (K=2, K=3).
    for (int r = tid; r < n; r += blockDim.x) {
        rowFrag[4 * r + 2] = rlw[r] + rsq[r] + rf[r] * IEL;   // u2_r (log2 dom)
        rowFrag[4 * r + 3] = 1.0f;
    }
    for (int j = tid; j < m; j += blockDim.x) {
        colFrag[4 * j + 2] = 1.0f;
        colFrag[4 * j + 3] = csq[j];                          // v2_j (log2 dom)
    }
    __syncthreads();

    const v2f* rowV = (const v2f*)rowFrag;
    const v2f* colV = (const v2f*)colFrag;
    const int lane   = tid & 31;
    const int wave   = tid >> 5;
    const int nWaves = blockDim.x >> 5;
    const int l15    = lane & 15;
    const int hi     = lane >> 4;      // 0 -> K0,K1 half; 1 -> K2,K3 half
    const int tilesI = n >> 4;         // multiple of 4 (16 or 64)
    const int tilesJ = m >> 4;

    for (int jt = wave; jt < tilesJ; jt += nWaves) {   // wave-uniform bounds
        const int j = (jt << 4) + l15;
        const v2f bfrag = colV[2 * j + hi];            // one ds_load_b64

        float mRun = -3.0e38f;                          // online softmax state
        float sRun = 0.0f;

        // 4 tiles (64 rows) per online-softmax update.
        for (int it0 = 0; it0 < tilesI; it0 += 4) {
            const int rb = (it0 << 4) + l15;
            const v2f a0 = rowV[2 * rb + hi];           // 4x ds_load_b64
            const v2f a1 = rowV[2 * (rb + 16) + hi];    // (pairs -> 2addr)
            const v2f a2 = rowV[2 * (rb + 32) + hi];
            const v2f a3 = rowV[2 * (rb + 48) + hi];

            const v8f z = {0.f, 0.f, 0.f, 0.f, 0.f, 0.f, 0.f, 0.f};
            v8f d0 = __builtin_amdgcn_wmma_f32_16x16x4_f32(
                false, a0, false, bfrag, (short)0, z, false, false);
            v8f d1 = __builtin_amdgcn_wmma_f32_16x16x4_f32(
                false, a1, false, bfrag, (short)0, z, false, false);
            v8f d2 = __builtin_amdgcn_wmma_f32_16x16x4_f32(
                false, a2, false, bfrag, (short)0, z, false, false);
            v8f d3 = __builtin_amdgcn_wmma_f32_16x16x4_f32(
                false, a3, false, bfrag, (short)0, z, false, false);

            const float bm = fmaxf(fmaxf(max8(d0), max8(d1)),
                                   fmaxf(max8(d2), max8(d3)));
            const float mNew = fmaxf(mRun, bm);
            // 32 independent raw exp2s + balanced-tree sum
            const float s0 = expsum8(d0, mNew);
            const float s1 = expsum8(d1, mNew);
            const float s2 = expsum8(d2, mNew);
            const float s3 = expsum8(d3, mNew);
            sRun = sRun * __builtin_amdgcn_exp2f(mRun - mNew)
                 + ((s0 + s1) + (s2 + s3));
            mRun = mNew;
        }
        // lanes l and l^16 hold rows 0-7 / 8-15 of the same column: merge
        float mO = __shfl_xor(mRun, 16);
        float sO = __shfl_xor(sRun, 16);
        float M  = fmaxf(mRun, mO);
        float S  = sRun * __builtin_amdgcn_exp2f(mRun - M)
                 + sO   * __builtin_amdgcn_exp2f(mO - M);
        float pot = -EPSLN2 * (M + __builtin_amdgcn_logf(S));  // back to natural
        if (lane < 16) outPot[j] = pot;
    }
    __syncthreads();   // outputs + fragments safe before next role refresh
}

__global__ __launch_bounds__(1024) void ot_sinkhorn_kernel(
    const float* __restrict__ dens,   // [B,1,32,32]
    const float* __restrict__ pts,    // [B,256,2]
    float* __restrict__ out)          // [1]
{
    __shared__ Smem sm;
    const int prob  = blockIdx.x;         // 24 problems: (batch, kind)
    const int batch = prob / 3;
    const int kind  = prob - 3 * batch;   // 0=ab, 1=aa, 2=bb
    const int tid   = threadIdx.x;
    const int nThr  = blockDim.x;

    int n, m;
    if      (kind == 0) { n = NMAX; m = PPTS; }
    else if (kind == 1) { n = NMAX; m = NMAX; }
    else                { n = PPTS; m = PPTS; }

    const float logInvP2 = -8.0f;   // log2(1/256), exact

    // ---- fill X side (rows of C) ----
    for (int i = tid; i < n; i += nThr) {
        float p0, p1, lw2;
        if (kind == 2) {                               // bb: gt points
            p0 = pts[(batch * PPTS + i) * 2 + 0];
            p1 = pts[(batch * PPTS + i) * 2 + 1];
            lw2 = logInvP2;
        } else {                                       // ab/aa: grid + density
            p0 = (float)((i & 31) * 8 + 4);            // cood = (c[k%32], c[k/32])
            p1 = (float)((i >> 5) * 8 + 4);
            float a = dens[batch * NMAX + i];
            lw2 = (a > 0.0f) ? __builtin_amdgcn_logf(a) : NEG_BIG;  // log2(a)
        }
        const float c0 = p0 - 128.0f, c1 = p1 - 128.0f;  // center: C invariant
        sm.fragX[4 * i + 0] = c0 * IEL;
        sm.fragX[4 * i + 1] = c1 * IEL;
        sm.lwX[i] = lw2;
        sm.sqX[i] = -0.5f * (c0 * c0 + c1 * c1) * IEL;
        sm.f[i]   = 0.0f;
    }
    // ---- fill Y side (cols of C) ----
    for (int j = tid; j < m; j += nThr) {
        float p0, p1, lw2;
        if (kind == 1) {                               // aa: grid again
            p0 = (float)((j & 31) * 8 + 4);
            p1 = (float)((j >> 5) * 8 + 4);
            float a = dens[batch * NMAX + j];
            lw2 = (a > 0.0f) ? __builtin_amdgcn_logf(a) : NEG_BIG;
        } else {                                       // ab/bb: gt points
            p0 = pts[(batch * PPTS + j) * 2 + 0];
            p1 = pts[(batch * PPTS + j) * 2 + 1];
            lw2 = logInvP2;
        }
        const float c0 = p0 - 128.0f, c1 = p1 - 128.0f;
        sm.fragY[4 * j + 0] = c0;                      // raw (X side carries scale)
        sm.fragY[4 * j + 1] = c1;
        sm.lwY[j] = lw2;
        sm.sqY[j] = -0.5f * (c0 * c0 + c1 * c1) * IEL;
        sm.g[j]   = 0.0f;
    }
    __syncthreads();

    // ---- 100 Sinkhorn iterations (g then f, detached weights) ----
    for (int iter = 0; iter < ITERS_; ++iter) {
        half_step(sm.fragX, sm.lwX, sm.sqX, sm.f,
                  sm.fragY, sm.sqY, sm.g, n, m);
        half_step(sm.fragY, sm.lwY, sm.sqY, sm.g,
                  sm.fragX, sm.sqX, sm.f, m, n);
    }
    // ---- final extrapolation: g_fin from loop f; f_fin from loop g ----
    half_step(sm.fragX, sm.lwX, sm.sqX, sm.f,
              sm.fragY, sm.sqY, sm.gfin, n, m);
    half_step(sm.fragY, sm.lwY, sm.sqY, sm.g,
              sm.fragX, sm.sqX, sm.f, m, n);          // overwrites f with f_fin

    // ---- dual value <a, f_fin> + <b, g_fin>  (weights = 2^lw2) ----
    float part = 0.0f;
    for (int i = tid; i < n; i += nThr)
        part += __builtin_amdgcn_exp2f(sm.lwX[i]) * sm.f[i];
    for (int j = tid; j < m; j += nThr)
        part += __builtin_amdgcn_exp2f(sm.lwY[j]) * sm.gfin[j];

    for (int off = 16; off > 0; off >>= 1) part += __shfl_down(part, off);
    const int lane = tid & 31, wave = tid >> 5;
    if (lane == 0) sm.red[wave] = part;
    __syncthreads();
    if (wave == 0) {
        const int nW = nThr >> 5;
        float v = (lane < nW) ? sm.red[lane] : 0.0f;
        for (int off = 16; off > 0; off >>= 1) v += __shfl_down(v, off);
        if (lane == 0) {
            // loss = ab - 0.5*(aa + bb), summed over batches
            float sign = (kind == 0) ? 1.0f : -0.5f;
            atomicAdd(out, sign * v);
        }
    }
}

extern "C" void kernel_launch(void* const* d_in, const int* in_sizes, int n_in,
                              void* d_out, int out_size, void* d_ws, size_t ws_size,
                              hipStream_t stream) {
    const float* dens = (const float*)d_in[0];   // [8,1,32,32] f32
    const float* pts  = (const float*)d_in[1];   // [8,256,2]   f32
    float* out = (float*)d_out;                  // [1]         f32

    hipMemsetAsync(out, 0, sizeof(float) * (size_t)out_size, stream);
    ot_sinkhorn_kernel<<<dim3(24), dim3(1024), 0, stream>>>(dens, pts, out);
}